// MN_neuron_braille_trained_90615220011498
// MI455X (gfx1250) — compile-verified
//
#include <hip/hip_runtime.h>
#include <cstdint>
#include <cstddef>

// ---------------------------------------------------------------------------
// MN-neuron scan, MI455X (gfx1250).
// Memory-bound streaming recurrence (262 MB @ 23.3 TB/s => ~11 us floor).
// Deep async global->LDS prefetch pipeline (CDNA5 GLOBAL_LOAD_ASYNC_TO_LDS_B128
// + ASYNCcnt) keeps ~16 MB in flight (matches BW*latency product) despite only
// 32K parallel streams. No matmul in this op => no WMMA by design.
// ---------------------------------------------------------------------------

#define T_STEPS   1000
#define BATCH     64
#define NNEUR     512
#define BN        (BATCH * NNEUR)      // 32768 independent streams
#define BLOCK     128                  // 4 wave32 per workgroup
#define NBLOCKS   (BN / BLOCK)         // 256 workgroups -> spread across WGPs
#define STAGES    128                  // LDS ring depth in timesteps (pow2)
#define GRP       4                    // timesteps covered per async b128 op
#define NGROUPS   (T_STEPS / GRP)      // 250
#define AHEAD     (STAGES / GRP)       // 32 outstanding async ops per wave
#define STEADY    (NGROUPS - AHEAD)    // 218 steady-state groups
#define STAGE_BYTES   (BLOCK * 4)      // 512 B per stage
#define STAGE_SHIFT   9                // log2(STAGE_BYTES)
#define ELEM_SHIFT    7                // log2(BLOCK) for float indexing
#define GROUP_GBYTES  ((size_t)GRP * BN * 4)   // global advance per group

extern __shared__ float lds_ring[];    // STAGES * BLOCK floats = 64 KB

// One wave32 instruction: copies 512B (4 timesteps x 32 elements, 16B/lane)
// from global memory into the LDS ring. Tracked by ASYNCcnt; async loads from
// one wave complete in order.
__device__ __forceinline__ void async_load_group(uint64_t gaddr, uint32_t ldsaddr) {
    asm volatile("global_load_async_to_lds_b128 %0, %1, off"
                 :: "v"(ldsaddr), "v"(gaddr)
                 : "memory");
}

__device__ __forceinline__ void wait_async_le_ahead_m1() {
    // exactly AHEAD loads outstanding-issued here in steady state, so
    // <= AHEAD-1 remaining  =>  the oldest (this group's) load is complete
    asm volatile("s_wait_asynccnt 31" ::: "memory");
}

__device__ __forceinline__ void wait_async_drain() {
    asm volatile("s_wait_asynccnt 0" ::: "memory");
}

__device__ __forceinline__ void wait_ds_done() {
    // close LDS WAR hazard before the async engine overwrites a retired stage
    asm volatile("s_wait_dscnt 0" ::: "memory");
}

__global__ __launch_bounds__(BLOCK) void mn_neuron_scan_kernel(
    const float* __restrict__ x,      // [T, B, N]
    const float* __restrict__ linear, // [N]
    const float* __restrict__ a,      // [N]
    const float* __restrict__ A1,     // [N]
    const float* __restrict__ A2,     // [N]
    float* __restrict__ out)          // [T, B, N] spikes
{
    // ---- model constants (folded at compile time) ----
    constexpr float DT   = 0.01f;
    constexpr float EL   = -0.07f, VR = -0.07f, TR = -0.06f, TINF = -0.05f;
    constexpr float BV   = 12.77495288848877f;
    constexpr float GV   = 45.24007797241211f;
    constexpr float K1   = 200.0f, K2 = 20.0f;
    constexpr float R1   = 0.3858567178249359f, R2 = -1.1421641111373901f;
    constexpr float d1   = 1.0f - K1 * DT;        // = -1.0f exactly
    constexpr float d2   = 1.0f - K2 * DT;        // =  0.8f
    constexpr float vc   = 1.0f - GV * DT;        // V decay
    constexpr float vk   = DT * GV * EL;          // V leak term
    constexpr float tc   = 1.0f - BV * DT;        // Thr decay
    constexpr float tk   = DT * BV * TINF;        // Thr bias term

    const int tid = threadIdx.x;
    const int gid = blockIdx.x * BLOCK + tid;     // element id in [0, BN)
    const int n   = gid & (NNEUR - 1);            // neuron index (N = 512 pow2)

    // per-neuron parameters (one load each, live in VGPRs for all 1000 steps)
    const float lin = linear[n];
    const float ak  = DT * a[n];
    const float a1v = A1[n];
    const float a2v = A2[n];

    // ---- async-copy lane geometry ----
    const int wave = tid >> 5;                    // 0..3
    const int lane = tid & 31;
    const int tsub = lane >> 3;                   // timestep within group: 0..3
    const int seg  = lane & 7;                    // 16B segment: 0..7
    const uint32_t lds_laneoff = (uint32_t)(wave * 128 + seg * 16);
    const uint32_t lds_base    = (uint32_t)(uintptr_t)&lds_ring[0];
    // per-lane global byte address of this wave's 128B slice at timestep tsub
    uint64_t ga = (uint64_t)(uintptr_t)x
                + (uint64_t)(blockIdx.x * BLOCK + wave * 32 + seg * 4) * 4u
                + (uint64_t)tsub * (BN * 4u);

    // ---- prologue: fill the pipeline (AHEAD groups = 128 timesteps) ----
    #pragma unroll 4
    for (int p = 0; p < AHEAD; ++p) {
        const uint32_t stage = (uint32_t)(p * GRP) & (STAGES - 1);
        async_load_group(ga,
            lds_base + ((stage + (uint32_t)tsub) << STAGE_SHIFT) + lds_laneoff);
        ga += GROUP_GBYTES;
    }

    // ---- per-thread recurrent state ----
    float V = EL, i1 = 0.0f, i2 = 0.0f, Thr = TINF;
    float* op = out + gid;                        // advances by BN per step

    #define MN_STEP(tt)                                                        \
        do {                                                                   \
            const float xt =                                                   \
                lds_ring[(((tt) & (STAGES - 1)) << ELEM_SHIFT) + tid];         \
            i1 *= d1;                                                          \
            i2 *= d2;                                                          \
            const float inp = fmaf(lin, xt, i1 + i2);                          \
            V   = fmaf(vc, V, fmaf(DT, inp, vk));                              \
            Thr = fmaf(tc, Thr, fmaf(ak, V - EL, tk));                         \
            const bool  s   = V > Thr;                                         \
            const float spk = s ? 1.0f : 0.0f;                                 \
            i1  = s ? fmaf(R1, i1, a1v) : i1;                                  \
            i2  = s ? fmaf(R2, i2, a2v) : i2;                                  \
            Thr = s ? fmaxf(Thr, TR)    : Thr;                                 \
            V   = s ? VR                : V;                                   \
            __builtin_nontemporal_store(spk, op);  /* write-once data */       \
            op += BN;                                                          \
        } while (0)

    // ---- steady state: exactly AHEAD loads in flight at every wait ----
    for (int g = 0; g < STEADY; ++g) {
        wait_async_le_ahead_m1();                 // group g resident in LDS
        #pragma unroll
        for (int k = 0; k < GRP; ++k) MN_STEP(g * GRP + k);

        wait_ds_done();                           // group g's stages retired
        const uint32_t stage = (uint32_t)((g + AHEAD) * GRP) & (STAGES - 1);
        async_load_group(ga,
            lds_base + ((stage + (uint32_t)tsub) << STAGE_SHIFT) + lds_laneoff);
        ga += GROUP_GBYTES;
    }

    // ---- tail: no more issues, so per-group "<= AHEAD-1" waits would be
    // vacuous; drain once (these loads were issued >= AHEAD groups ago),
    // then consume the last AHEAD groups from LDS with pure compute ----
    wait_async_drain();
    for (int g = STEADY; g < NGROUPS; ++g) {
        #pragma unroll
        for (int k = 0; k < GRP; ++k) MN_STEP(g * GRP + k);
    }
    #undef MN_STEP
}

extern "C" void kernel_launch(void* const* d_in, const int* in_sizes, int n_in,
                              void* d_out, int out_size, void* d_ws, size_t ws_size,
                              hipStream_t stream) {
    (void)in_sizes; (void)n_in; (void)out_size; (void)d_ws; (void)ws_size;
    const float* x      = (const float*)d_in[0];
    const float* linear = (const float*)d_in[1];
    const float* a      = (const float*)d_in[2];
    const float* A1     = (const float*)d_in[3];
    const float* A2     = (const float*)d_in[4];
    float* out          = (float*)d_out;

    const size_t shmem = (size_t)STAGES * STAGE_BYTES;  // 64 KB of 320 KB/WGP
    mn_neuron_scan_kernel<<<NBLOCKS, BLOCK, shmem, stream>>>(x, linear, a, A1, A2, out);
}